// XConv_2121713844234
// MI455X (gfx1250) — compile-verified
//
#include <hip/hip_runtime.h>
#include <hip/hip_bf16.h>

typedef float v2f __attribute__((ext_vector_type(2)));
typedef float v8f __attribute__((ext_vector_type(8)));

#define BDIM 4
#define CDIM 64
#define NPTS 4096
#define KNN  16
#define OUTC 64
#define NEG_INF (-3.402823466e38f)

// Async global->LDS gather (CDNA5 GLOBAL_LOAD_ASYNC_TO_LDS_B32, ASYNCcnt).
#define USE_ASYNC_GATHER 1

__device__ __forceinline__ v8f wmma4(v2f a, v2f b, v8f c) {
  // V_WMMA_F32_16X16X4_F32 : D = A(16x4) * B(4x16) + C(16x16)
  return __builtin_amdgcn_wmma_f32_16x16x4_f32(false, a, false, b, (short)0, c,
                                               false, false);
}

// ---------------------------------------------------------------- norms
__global__ void norms_kernel(const float* __restrict__ x,
                             float* __restrict__ xx, float* __restrict__ xx3) {
  int g = blockIdx.x * blockDim.x + threadIdx.x;
  if (g >= BDIM * NPTS) return;
  int b = g / NPTS, n = g % NPTS;
  const float* xb = x + (size_t)b * CDIM * NPTS;
  float s = 0.f, s3 = 0.f;
  for (int c = 0; c < CDIM; ++c) {
    float v = xb[(size_t)c * NPTS + n];
    s += v * v;
    if (c < 3) s3 += v * v;
  }
  xx[g] = s;
  xx3[g] = s3;
}

// ------------------------------------------------------- knn (WMMA f32 gram)
// One workgroup = one 16-row tile of pd; 16 x 4096 f32 rows (256KB) + 8 x 4KB
// per-wave selection scratch in LDS (295KB of the 320KB/WGP).
template <int CD>
__global__ void knn_kernel(const float* __restrict__ x,
                           const float* __restrict__ xx,
                           int* __restrict__ idx_out) {
  extern __shared__ float pd[];  // [16][NPTS] then per-wave scratch
  const int lane = threadIdx.x & 31;
  const int wv = threadIdx.x >> 5;
  const int nwaves = blockDim.x >> 5;
  const int half = lane >> 4;
  const int l16 = lane & 15;
  const int b = blockIdx.x / (NPTS / 16);
  const int row0 = (blockIdx.x % (NPTS / 16)) * 16;
  const float* xb = x + (size_t)b * CDIM * NPTS;
  const float* xxb = xx + (size_t)b * NPTS;

  constexpr int KS = (CD + 3) / 4;  // 16 for CD=64, 1 for CD=3 (zero padded)

  // A fragments: A(m,k) = x[b][k][row0+m]; lane<16: K=4kk+{0,1}, lane>=16: +{2,3}
  v2f af[KS];
#pragma unroll
  for (int kk = 0; kk < KS; ++kk) {
    int k0 = 4 * kk + 2 * half, k1 = k0 + 1;
    v2f a;
    a.x = (k0 < CD) ? xb[(size_t)k0 * NPTS + row0 + l16] : 0.f;
    a.y = (k1 < CD) ? xb[(size_t)k1 * NPTS + row0 + l16] : 0.f;
    af[kk] = a;
  }
  float rn[8];
#pragma unroll
  for (int r = 0; r < 8; ++r) rn[r] = xxb[row0 + r + 8 * half];

  for (int ct = wv; ct < NPTS / 16; ct += nwaves) {
    const int ncol = ct * 16 + l16;
    __builtin_prefetch(&xb[ncol + nwaves * 16], 0, 1);
    v8f acc = {};
#pragma unroll
    for (int kk = 0; kk < KS; ++kk) {
      int k0 = 4 * kk + 2 * half, k1 = k0 + 1;
      v2f bf;
      bf.x = (k0 < CD) ? xb[(size_t)k0 * NPTS + ncol] : 0.f;
      bf.y = (k1 < CD) ? xb[(size_t)k1 * NPTS + ncol] : 0.f;
      acc = wmma4(af[kk], bf, acc);
    }
    const float cn = xxb[ncol];
#pragma unroll
    for (int r = 0; r < 8; ++r) {
      int M = r + 8 * half;
      pd[M * NPTS + ncol] = 2.0f * acc[r] - rn[r] - cn;  // -||xi-xj||^2
    }
  }
  __syncthreads();

  // ---- top-16 per row: single-pass per-lane sorted top-16 + butterfly merge
  float* scr = pd + 16 * NPTS + wv * 1024;  // per-wave: 512 vals + 512 idx
  float* lv = scr;
  int* li = (int*)(scr + 512);

  for (int rr = 0; rr < 2; ++rr) {
    const int row = 2 * wv + rr;
    const float* prow = pd + row * NPTS;
    int* orow = idx_out + ((size_t)(b * NPTS + row0 + row)) * KNN;

#pragma unroll
    for (int i = 0; i < 16; ++i) {
      lv[i * 32 + lane] = NEG_INF;
      li[i * 32 + lane] = 0x7fffffff;
    }
    float minv = NEG_INF;
    // phase 1: each lane builds sorted-desc top-16 of its 128 elements
    for (int c = lane; c < NPTS; c += 32) {
      float v = prow[c];
      if (v > minv) {
        int pos = 15;
        while (pos > 0 && lv[(pos - 1) * 32 + lane] < v) {
          lv[pos * 32 + lane] = lv[(pos - 1) * 32 + lane];
          li[pos * 32 + lane] = li[(pos - 1) * 32 + lane];
          --pos;
        }
        lv[pos * 32 + lane] = v;
        li[pos * 32 + lane] = c;
        minv = lv[15 * 32 + lane];
      }
    }
    // phase 2: 16-way merge of 32 sorted streams via shfl_xor reduction
    int ptr = 0;
    for (int j = 0; j < KNN; ++j) {
      float bv = (ptr < 16) ? lv[ptr * 32 + lane] : NEG_INF;
      int bi = (ptr < 16) ? li[ptr * 32 + lane] : 0x7fffffff;
      float mv = bv;
      int mi = bi;
#pragma unroll
      for (int s = 16; s >= 1; s >>= 1) {
        float ov = __shfl_xor(mv, s, 32);
        int oi = __shfl_xor(mi, s, 32);
        if (ov > mv || (ov == mv && oi < mi)) { mv = ov; mi = oi; }
      }
      if (lane == 0) orow[j] = mi;
      if (bi == mi) ++ptr;  // winner lane advances its stream
    }
  }
}

// ------------------------------------------------- wbar = mean_l w_xt2 fold
__global__ void wbar_kernel(const float* __restrict__ w_xt2,
                            float* __restrict__ wbar) {
  int t = threadIdx.x;  // 256
  int j = t >> 4, i = t & 15;
  float s = 0.f;
  for (int l = 0; l < 16; ++l) s += w_xt2[(j * 16 + l) * 16 + i];
  wbar[j * 16 + i] = s * (1.0f / 16.0f);
}

// ------------------------------------------------- z1 BN statistics pass
__global__ void z1_stats_kernel(const float* __restrict__ x,
                                const int* __restrict__ idx3,
                                const float* __restrict__ w_xt1,
                                float* __restrict__ stats1) {
  __shared__ float s1[32];
  if (threadIdx.x < 32) s1[threadIdx.x] = 0.f;
  __syncthreads();
  int g = blockIdx.x * blockDim.x + threadIdx.x;
  int b = g / NPTS, n = g % NPTS;
  const float* xb = x + (size_t)b * CDIM * NPTS;
  const int* i3 = idx3 + (size_t)g * KNN;
  float cx0 = xb[n], cx1 = xb[NPTS + n], cx2 = xb[2 * NPTS + n];
  float ls[16], lq[16];
#pragma unroll
  for (int o = 0; o < 16; ++o) { ls[o] = 0.f; lq[o] = 0.f; }
  for (int r = 0; r < KNN; ++r) {
    int nb = i3[r];
    float f0 = xb[nb] - cx0, f1 = xb[NPTS + nb] - cx1, f2 = xb[2 * NPTS + nb] - cx2;
#pragma unroll
    for (int o = 0; o < 16; ++o) {
      float z = f0 * w_xt1[o * 3] + f1 * w_xt1[o * 3 + 1] + f2 * w_xt1[o * 3 + 2];
      ls[o] += z; lq[o] += z * z;
    }
  }
#pragma unroll
  for (int o = 0; o < 16; ++o) {
    atomicAdd(&s1[o], ls[o]);
    atomicAdd(&s1[16 + o], lq[o]);
  }
  __syncthreads();
  if (threadIdx.x < 32) atomicAdd(&stats1[threadIdx.x], s1[threadIdx.x]);
}

// ------------------------------------------------- BN finalize (mean/invstd)
__global__ void finalize_kernel(const float* __restrict__ stats,
                                float* __restrict__ bn, int nch, float invc) {
  int c = threadIdx.x;
  if (c < nch) {
    float mean = stats[c] * invc;
    float var = stats[nch + c] * invc - mean * mean;
    bn[c] = mean;
    bn[nch + c] = rsqrtf(var + 1e-5f);
  }
}

// ------------------------------------------------- main per-point WMMA chain
// one wave per point: z1 -> X ; async-gather neigh -> G = feat_x @ w_c1^T ;
// z2 = X @ G.  feat_x cols [64,128) are the broadcast center: never gathered,
// folded into the A-fragment read (row-independent).
#define WV_FLOATS 2624  // 256 z1s + 256 Xs + 1024 fxs + 1024 Gs + 64 cens
__global__ void main_kernel(const float* __restrict__ x,
                            const int* __restrict__ idx,
                            const int* __restrict__ idx3,
                            const float* __restrict__ w_xt1,
                            const float* __restrict__ bn1,
                            const float* __restrict__ g1,
                            const float* __restrict__ b1,
                            const float* __restrict__ wbar,
                            const float* __restrict__ w_c1,
                            float* __restrict__ z2buf,
                            float* __restrict__ stats2) {
  extern __shared__ float sm[];
  const int lane = threadIdx.x & 31;
  const int wv = threadIdx.x >> 5;
  const int half = lane >> 4;
  const int l16 = lane & 15;
  float* z1s = sm + wv * WV_FLOATS;    // 16x16
  float* Xs = z1s + 256;               // 16x16
  float* fxs = Xs + 256;               // 16x64 raw gathered neighbors
  float* Gs = fxs + 1024;              // 16x64
  float* cens = Gs + 1024;             // 64 center features
  float* smst = sm + 4 * WV_FLOATS;    // 128 stats
  for (int t = threadIdx.x; t < 128; t += blockDim.x) smst[t] = 0.f;
  __syncthreads();

  const int p = blockIdx.x * (blockDim.x >> 5) + wv;
  const int b = p / NPTS, n = p % NPTS;
  const float* xb = x + (size_t)b * CDIM * NPTS;

  // ---- kick off async gather of 16x64 neighbor block early
  {
    const int* ir = idx + (size_t)p * KNN;
#if USE_ASYNC_GATHER
#pragma unroll 4
    for (int q = 0; q < 32; ++q) {
      int t = lane + 32 * q, j = t >> 6, c = t & 63;
      int nb = ir[j];
      const float* src = xb + (size_t)c * NPTS + nb;
      unsigned ldsa = (unsigned)(uintptr_t)(const void*)&fxs[j * 64 + c];
      asm volatile("global_load_async_to_lds_b32 %0, %1, off"
                   :
                   : "v"(ldsa), "v"((unsigned long long)(uintptr_t)src)
                   : "memory");
    }
#else
#pragma unroll 4
    for (int q = 0; q < 32; ++q) {
      int t = lane + 32 * q, j = t >> 6, c = t & 63;
      int nb = ir[j];
      fxs[j * 64 + c] = xb[(size_t)c * NPTS + nb];
    }
#endif
    for (int c = lane; c < 64; c += 32) cens[c] = xb[(size_t)c * NPTS + n];
  }

  // ---- z1 = relu(bn1(feat_xyz @ w_xt1^T)), staged A-major in LDS
  {
    const int* i3 = idx3 + (size_t)p * KNN;
#pragma unroll
    for (int q = 0; q < 8; ++q) {
      int t = lane + 32 * q, r = t >> 4, o = t & 15;
      int nb = i3[r];
      float z = 0.f;
#pragma unroll
      for (int c = 0; c < 3; ++c)
        z += (xb[(size_t)c * NPTS + nb] - xb[(size_t)c * NPTS + n]) * w_xt1[o * 3 + c];
      z = (z - bn1[o]) * bn1[16 + o] * g1[o] + b1[o];
      z1s[r * 16 + o] = fmaxf(z, 0.f);
    }
  }

  // ---- X = z1 @ wbar^T  (16x16, K=16 -> 4 wmma steps)
  {
    v8f aX = {};
#pragma unroll
    for (int kk = 0; kk < 4; ++kk) {
      int k0 = 4 * kk + 2 * half, k1 = k0 + 1;
      v2f a; a.x = z1s[l16 * 16 + k0]; a.y = z1s[l16 * 16 + k1];
      v2f bf; bf.x = wbar[l16 * 16 + k0]; bf.y = wbar[l16 * 16 + k1];
      aX = wmma4(a, bf, aX);
    }
#pragma unroll
    for (int r = 0; r < 8; ++r) Xs[(r + 8 * half) * 16 + l16] = aX[r];
  }

#if USE_ASYNC_GATHER
  asm volatile("s_wait_asynccnt 0x0" ::: "memory");
#endif

  // ---- G = feat_x @ w_c1^T  (16x64, K=128 -> 32 wmma steps x 4 tiles)
  {
    v8f g4[4];
#pragma unroll
    for (int t0 = 0; t0 < 4; ++t0) g4[t0] = (v8f){};
    // k in [0,64): A = gathered neighbor - center
#pragma unroll 4
    for (int kk = 0; kk < 16; ++kk) {
      int k0 = 4 * kk + 2 * half, k1 = k0 + 1;
      v2f a;
      a.x = fxs[l16 * 64 + k0] - cens[k0];
      a.y = fxs[l16 * 64 + k1] - cens[k1];
#pragma unroll
      for (int t0 = 0; t0 < 4; ++t0) {
        int o = t0 * 16 + l16;
        v2f bf; bf.x = w_c1[o * 128 + k0]; bf.y = w_c1[o * 128 + k1];
        g4[t0] = wmma4(a, bf, g4[t0]);
      }
    }
    // k in [64,128): A = broadcast center (row-independent)
#pragma unroll 4
    for (int kk = 16; kk < 32; ++kk) {
      int k0 = 4 * kk + 2 * half, k1 = k0 + 1;
      v2f a; a.x = cens[k0 - 64]; a.y = cens[k1 - 64];
#pragma unroll
      for (int t0 = 0; t0 < 4; ++t0) {
        int o = t0 * 16 + l16;
        v2f bf; bf.x = w_c1[o * 128 + k0]; bf.y = w_c1[o * 128 + k1];
        g4[t0] = wmma4(a, bf, g4[t0]);
      }
    }
#pragma unroll
    for (int t0 = 0; t0 < 4; ++t0)
#pragma unroll
      for (int r = 0; r < 8; ++r)
        Gs[(r + 8 * half) * 64 + t0 * 16 + l16] = g4[t0][r];
  }

  // ---- z2 = X @ G  (16x64, K=16), store pre-BN + accumulate stats
  {
    v8f z4[4];
#pragma unroll
    for (int t0 = 0; t0 < 4; ++t0) z4[t0] = (v8f){};
#pragma unroll
    for (int kk = 0; kk < 4; ++kk) {
      int k0 = 4 * kk + 2 * half, k1 = k0 + 1;
      v2f a; a.x = Xs[l16 * 16 + k0]; a.y = Xs[l16 * 16 + k1];
#pragma unroll
      for (int t0 = 0; t0 < 4; ++t0) {
        v2f bf; bf.x = Gs[k0 * 64 + t0 * 16 + l16]; bf.y = Gs[k1 * 64 + t0 * 16 + l16];
        z4[t0] = wmma4(a, bf, z4[t0]);
      }
    }
    float* zp = z2buf + (size_t)p * (KNN * OUTC);
#pragma unroll
    for (int t0 = 0; t0 < 4; ++t0) {
      int o = t0 * 16 + l16;
      float s = 0.f, sq = 0.f;
#pragma unroll
      for (int r = 0; r < 8; ++r) {
        float v = z4[t0][r];
        zp[(r + 8 * half) * OUTC + o] = v;
        s += v; sq += v * v;
      }
      atomicAdd(&smst[o], s);
      atomicAdd(&smst[64 + o], sq);
    }
  }
  __syncthreads();
  if (threadIdx.x < 128) atomicAdd(&stats2[threadIdx.x], smst[threadIdx.x]);
}

// ------------------------------------------------- z3 = relu(bn2(z2)) @ w_c2^T
__global__ void z3_kernel(const float* __restrict__ z2buf,
                          const float* __restrict__ bn2,
                          const float* __restrict__ g2,
                          const float* __restrict__ b2,
                          const float* __restrict__ w_c2,
                          float* __restrict__ z3buf,
                          float* __restrict__ stats3) {
  extern __shared__ float sm[];
  const int lane = threadIdx.x & 31;
  const int wv = threadIdx.x >> 5;
  const int half = lane >> 4;
  const int l16 = lane & 15;
  float* h1s = sm + wv * 1024;   // 16x64
  float* smst = sm + 4 * 1024;   // 128 stats
  for (int t = threadIdx.x; t < 128; t += blockDim.x) smst[t] = 0.f;
  __syncthreads();

  const int p = blockIdx.x * (blockDim.x >> 5) + wv;
  const float* zp = z2buf + (size_t)p * (KNN * OUTC);
#pragma unroll 4
  for (int q = 0; q < 32; ++q) {
    int t = lane + 32 * q, c = t & 63;
    float v = zp[t];
    v = (v - bn2[c]) * bn2[64 + c] * g2[c] + b2[c];
    h1s[t] = fmaxf(v, 0.f);
  }

  v8f z4[4];
#pragma unroll
  for (int t0 = 0; t0 < 4; ++t0) z4[t0] = (v8f){};
#pragma unroll 4
  for (int kk = 0; kk < 16; ++kk) {
    int k0 = 4 * kk + 2 * half, k1 = k0 + 1;
    v2f a; a.x = h1s[l16 * 64 + k0]; a.y = h1s[l16 * 64 + k1];
#pragma unroll
    for (int t0 = 0; t0 < 4; ++t0) {
      int o = t0 * 16 + l16;
      v2f bf; bf.x = w_c2[o * 64 + k0]; bf.y = w_c2[o * 64 + k1];
      z4[t0] = wmma4(a, bf, z4[t0]);
    }
  }
  float* op = z3buf + (size_t)p * (KNN * OUTC);
#pragma unroll
  for (int t0 = 0; t0 < 4; ++t0) {
    int o = t0 * 16 + l16;
    float s = 0.f, sq = 0.f;
#pragma unroll
    for (int r = 0; r < 8; ++r) {
      float v = z4[t0][r];
      op[(r + 8 * half) * OUTC + o] = v;
      s += v; sq += v * v;
    }
    atomicAdd(&smst[o], s);
    atomicAdd(&smst[64 + o], sq);
  }
  __syncthreads();
  if (threadIdx.x < 128) atomicAdd(&stats3[threadIdx.x], smst[threadIdx.x]);
}

// ------------------------------------------------- epilogue: bn3+relu, max_k
__global__ void out_kernel(const float* __restrict__ z3buf,
                           const float* __restrict__ bn3,
                           const float* __restrict__ g3,
                           const float* __restrict__ b3,
                           float* __restrict__ out) {
  int g = blockIdx.x * blockDim.x + threadIdx.x;  // (b, o, n) flattened
  if (g >= BDIM * OUTC * NPTS) return;
  int b = g / (OUTC * NPTS);
  int o = (g / NPTS) % OUTC;
  int n = g % NPTS;
  const float* zp = z3buf + ((size_t)(b * NPTS + n) * KNN) * OUTC;
  float mean = bn3[o], is = bn3[64 + o], gg = g3[o], bb = b3[o];
  float best = NEG_INF;
  for (int r = 0; r < KNN; ++r) {
    float v = (zp[r * OUTC + o] - mean) * is * gg + bb;
    v = fmaxf(v, 0.f);
    best = fmaxf(best, v);
  }
  out[g] = best;
}

// ----------------------------------------------------------------- launcher
extern "C" void kernel_launch(void* const* d_in, const int* in_sizes, int n_in,
                              void* d_out, int out_size, void* d_ws, size_t ws_size,
                              hipStream_t stream) {
  const float* x     = (const float*)d_in[0];
  const float* w_xt1 = (const float*)d_in[1];
  const float* g_xt1 = (const float*)d_in[2];
  const float* b_xt1 = (const float*)d_in[3];
  const float* w_xt2 = (const float*)d_in[4];
  const float* w_c1  = (const float*)d_in[5];
  const float* g_c1  = (const float*)d_in[6];
  const float* b_c1  = (const float*)d_in[7];
  const float* w_c2  = (const float*)d_in[8];
  const float* g_c2  = (const float*)d_in[9];
  const float* b_c2  = (const float*)d_in[10];
  float* out = (float*)d_out;

  char* ws = (char*)d_ws;
  size_t off = 0;
  auto alloc = [&](size_t bytes) {
    char* p = ws + off;
    off += (bytes + 255) & ~(size_t)255;
    return p;
  };
  float* xx     = (float*)alloc(BDIM * NPTS * 4);
  float* xx3    = (float*)alloc(BDIM * NPTS * 4);
  int*   idx    = (int*)alloc((size_t)BDIM * NPTS * KNN * 4);
  int*   idx3   = (int*)alloc((size_t)BDIM * NPTS * KNN * 4);
  float* stats1 = (float*)alloc(32 * 4);
  float* bn1    = (float*)alloc(32 * 4);
  float* stats2 = (float*)alloc(128 * 4);
  float* bn2    = (float*)alloc(128 * 4);
  float* stats3 = (float*)alloc(128 * 4);
  float* bn3    = (float*)alloc(128 * 4);
  float* wbar   = (float*)alloc(256 * 4);
  float* z2buf  = (float*)alloc((size_t)BDIM * NPTS * KNN * OUTC * 4);
  float* z3buf  = (float*)alloc((size_t)BDIM * NPTS * KNN * OUTC * 4);

  hipMemsetAsync(stats1, 0, 32 * 4, stream);
  hipMemsetAsync(stats2, 0, 128 * 4, stream);
  hipMemsetAsync(stats3, 0, 128 * 4, stream);

  // pd rows (256KB) + 8 waves x 4KB selection scratch = 288KB+ (<=320KB/WGP)
  const size_t knn_lds = ((size_t)16 * NPTS + 8 * 1024) * 4;
  hipFuncSetAttribute((const void*)knn_kernel<CDIM>,
                      hipFuncAttributeMaxDynamicSharedMemorySize, (int)knn_lds);
  hipFuncSetAttribute((const void*)knn_kernel<3>,
                      hipFuncAttributeMaxDynamicSharedMemorySize, (int)knn_lds);

  const int npts_total = BDIM * NPTS;

  norms_kernel<<<npts_total / 256, 256, 0, stream>>>(x, xx, xx3);

  knn_kernel<CDIM><<<BDIM * (NPTS / 16), 256, knn_lds, stream>>>(x, xx, idx);
  knn_kernel<3><<<BDIM * (NPTS / 16), 256, knn_lds, stream>>>(x, xx3, idx3);

  wbar_kernel<<<1, 256, 0, stream>>>(w_xt2, wbar);
  z1_stats_kernel<<<npts_total / 256, 256, 0, stream>>>(x, idx3, w_xt1, stats1);

  const float invc = 1.0f / (float)((size_t)BDIM * NPTS * KNN);
  finalize_kernel<<<1, 64, 0, stream>>>(stats1, bn1, 16, invc);

  const size_t main_lds = (4 * WV_FLOATS + 128) * 4;
  main_kernel<<<npts_total / 4, 128, main_lds, stream>>>(
      x, idx, idx3, w_xt1, bn1, g_xt1, b_xt1, wbar, w_c1, z2buf, stats2);

  finalize_kernel<<<1, 64, 0, stream>>>(stats2, bn2, 64, invc);

  const size_t z3_lds = (4 * 1024 + 128) * 4;
  z3_kernel<<<npts_total / 4, 128, z3_lds, stream>>>(z2buf, bn2, g_c1, b_c1,
                                                     w_c2, z3buf, stats3);

  finalize_kernel<<<1, 64, 0, stream>>>(stats3, bn3, 64, invc);

  out_kernel<<<(BDIM * OUTC * NPTS) / 256, 256, 0, stream>>>(z3buf, bn3, g_c2,
                                                             b_c2, out);
}